// StgcnBlock_23235773071674
// MI455X (gfx1250) — compile-verified
//
#include <hip/hip_runtime.h>
#include <hip/hip_bf16.h>

// ---------------------------------------------------------------------------
// ST-GCN block on gfx1250 (MI455X), fp32 via V_WMMA_F32_16X16X4_F32.
// ~34 GFLOP fp32, ~300MB fused HBM traffic -> compute bound -> WMMA pipe.
// Deterministic: BN statistics via per-workgroup partial buffers reduced in a
// fixed order (no float atomics).
// ---------------------------------------------------------------------------

#define K_NUM  6
#define EDGE   3
#define LAMDA  0.5f
#define BN_EPS 1e-5f
#define NB     32      // batch
#define CIN    64
#define COUT   64
#define TT     300
#define VV     25
#define CT1    10      // t per workgroup (K1 and K3)
#define NCHUNK (TT / CT1)       // 30
#define NWG    (NB * NCHUNK)    // 960

typedef __attribute__((ext_vector_type(2))) float v2f;
typedef __attribute__((ext_vector_type(8))) float v8f;

// D = A(16x4) * B(4x16) + C  -- fp32 WMMA
__device__ __forceinline__ v8f wmma4(v2f a, v2f b, v8f c) {
  return __builtin_amdgcn_wmma_f32_16x16x4_f32(
      /*neg_a=*/false, a, /*neg_b=*/false, b,
      /*c_mod=*/(short)0, c, /*reuse_a=*/false, /*reuse_b=*/false);
}

// A-operand 16x4 tile from row-major LDS matrix [m][k], stride floats.
// A[m][k]: lane = m + 16*(k>>1), vgpr = k&1  ->  lane reads (m, keven..keven+1)
__device__ __forceinline__ v2f load_A(const float* base, int stride,
                                      int m_base, int k_base) {
  const int lane = threadIdx.x & 31;
  const int m = m_base + (lane & 15);
  const int k = k_base + ((lane >> 4) << 1);
  const float* p = base + m * stride + k;
  v2f r; r.x = p[0]; r.y = p[1];
  return r;
}

// B-operand 4x16 tile from row-major LDS matrix [k][n], stride floats.
// B[k][n]: vgpr j holds row k = j + 2*(lane>>4), col n = lane&15
__device__ __forceinline__ v2f load_B(const float* base, int stride,
                                      int k_base, int n_base) {
  const int lane = threadIdx.x & 31;
  const int n = n_base + (lane & 15);
  const int k = k_base + ((lane >> 4) << 1);
  const float* p = base + k * stride + n;
  v2f r; r.x = p[0]; r.y = p[stride];
  return r;
}

// ---------------------------------------------------------------------------
// K1: fused 1x1-conv GEMM + graph mixing.
//   Stage1: Y[o=0..383][v] = conv_w @ x_t   (+bias, v<25 only)
//   Stage2: G[c][w] = sum_k Y[k*64+c][:] @ M_k[:,w],  M_k = A_k | LAMDA*B[n,k]
// Writes g to ws and per-wave BN1 partial (sum, sumsq) to pbuf1.
// ---------------------------------------------------------------------------
__global__ __launch_bounds__(256)
void k1_gcn(const float* __restrict__ x, const float* __restrict__ Ag,
            const float* __restrict__ Bg, const float* __restrict__ cw,
            const float* __restrict__ cb, float* __restrict__ g_ws,
            float* __restrict__ pbuf1) {
  extern __shared__ float smem[];
  float* Wlds = smem;                  // [384][66]
  float* Ylds = Wlds + 384 * 66;       // [384][36]
  float* Mlds = Ylds + 384 * 36;       // [6][28][40]
  float* Xlds = Mlds + 6 * 28 * 40;    // [64][40]
  float* blds = Xlds + 64 * 40;        // [384]

  const int n    = blockIdx.x;
  const int t0   = blockIdx.y * CT1;
  const int tid  = threadIdx.x;
  const int wave = tid >> 5;
  const int lane = tid & 31;

  // Stage conv weights (natural A-operand layout: row-major [o][ci]).
  for (int i = tid; i < 384 * 64; i += 256)
    Wlds[(i >> 6) * 66 + (i & 63)] = cw[i];
  for (int i = tid; i < 384; i += 256) blds[i] = cb[i];
  // Stage per-n mixing matrices (B-operand layout [k][v][w], zero padded).
  for (int i = tid; i < 6 * 28 * 40; i += 256) {
    const int k = i / (28 * 40);
    const int r = i - k * 28 * 40;
    const int v = r / 40;
    const int w = r - v * 40;
    float val = 0.f;
    if (v < VV && w < VV)
      val = (k < EDGE) ? Ag[(k * VV + v) * VV + w]
                       : LAMDA * Bg[(((n * EDGE) + (k - EDGE)) * VV + v) * VV + w];
    Mlds[i] = val;
  }

  float bsum[8], bsq[8];
#pragma unroll
  for (int j = 0; j < 8; ++j) { bsum[j] = 0.f; bsq[j] = 0.f; }

  const int ct = wave >> 1;  // 0..3 : output channel tile
  const int wt = wave & 1;   // 0..1 : output vertex tile

  for (int t = t0; t < t0 + CT1; ++t) {
    __syncthreads();  // previous-iteration stage2 readers done
    // Stage x_t -> LDS (B-operand layout [ci][v], v>=25 zeroed).
    for (int i = tid; i < 64 * 40; i += 256) {
      const int ci = i / 40, v = i - ci * 40;
      Xlds[i] = (v < VV) ? x[((n * CIN + ci) * TT + t) * VV + v] : 0.f;
    }
    // Prefetch next t's activation rows into GL2 (global_prefetch_b8).
    if (t + 1 < TT)
      __builtin_prefetch(&x[((n * CIN + (tid & 63)) * TT + (t + 1)) * VV], 0, 3);
    __syncthreads();

    // Stage 1: Y(384x32) = W(384x64) @ X(64x32); 48 tiles over 8 waves.
    for (int tl = 0; tl < 6; ++tl) {
      const int tile = wave + tl * 8;  // 0..47
      const int ot = tile >> 1;        // 0..23
      const int vt = tile & 1;
      v8f acc = {};
#pragma unroll
      for (int s = 0; s < 16; ++s) {
        v2f a = load_A(Wlds, 66, ot * 16, s * 4);
        v2f b = load_B(Xlds, 40, s * 4, vt * 16);
        acc = wmma4(a, b, acc);
      }
      // +bias on real columns only (keeps v-pad exactly zero), store row-major.
      const int nn = vt * 16 + (lane & 15);
      const int mh = ot * 16 + ((lane >> 4) << 3);
#pragma unroll
      for (int j = 0; j < 8; ++j) {
        float val = acc[j];
        if (nn < VV) val += blds[mh + j];
        Ylds[(mh + j) * 36 + nn] = val;
      }
    }
    __syncthreads();

    // Stage 2: G(64x32) = sum_k Y_k @ M_k ; one 16x16 tile per wave.
    v8f acc = {};
#pragma unroll
    for (int k = 0; k < 6; ++k) {
#pragma unroll
      for (int s = 0; s < 7; ++s) {  // K = 28 (v padded 25->28, pad is zero)
        v2f a = load_A(Ylds, 36, k * 64 + ct * 16, s * 4);
        v2f b = load_B(Mlds + k * 28 * 40, 40, s * 4, wt * 16);
        acc = wmma4(a, b, acc);
      }
    }
    // Epilogue: store g, accumulate BN1 partials in registers.
    const int v  = wt * 16 + (lane & 15);
    const int mh = ct * 16 + ((lane >> 4) << 3);
    if (v < VV) {
#pragma unroll
      for (int j = 0; j < 8; ++j) {
        const float val = acc[j];
        g_ws[((n * COUT + (mh + j)) * TT + t) * VV + v] = val;
        bsum[j] += val;
        bsq[j]  += val * val;
      }
    }
  }

  // Reduce partials over the 16 lanes that share channels (xor within half).
#pragma unroll
  for (int j = 0; j < 8; ++j) {
#pragma unroll
    for (int m = 1; m < 16; m <<= 1) {
      bsum[j] += __shfl_xor(bsum[j], m, 32);
      bsq[j]  += __shfl_xor(bsq[j], m, 32);
    }
  }
  if ((lane & 15) == 0) {
    const int wg = blockIdx.y * gridDim.x + blockIdx.x;  // 0..959
    const int hi = lane >> 4;
#pragma unroll
    for (int j = 0; j < 8; ++j) {
      const int cl = hi * 8 + j;                   // local channel 0..15
      const int base = ((wg * 8 + wave) * 16 + cl) * 2;
      pbuf1[base + 0] = bsum[j];
      pbuf1[base + 1] = bsq[j];
    }
  }
}

// ---------------------------------------------------------------------------
// K2a: fold partials -> per-channel BN scale/shift (fixed order = determinism)
// ---------------------------------------------------------------------------
__global__ void k2_stats(const float* __restrict__ pbuf,
                         const float* __restrict__ gamma,
                         const float* __restrict__ beta,
                         float* __restrict__ stats) {
  const int c = threadIdx.x;
  if (c >= COUT) return;
  const int ct = c >> 4, cl = c & 15;
  float s = 0.f, q = 0.f;
  for (int wg = 0; wg < NWG; ++wg) {
#pragma unroll
    for (int w2 = 0; w2 < 2; ++w2) {
      const int base = ((wg * 8 + (ct * 2 + w2)) * 16 + cl) * 2;
      s += pbuf[base];
      q += pbuf[base + 1];
    }
  }
  const float inv  = 1.f / (float)(NB * TT * VV);
  const float mean = s * inv;
  const float var  = q * inv - mean * mean;
  const float sc   = gamma[c] * rsqrtf(var + BN_EPS);
  stats[c]        = sc;
  stats[COUT + c] = beta[c] - mean * sc;
}

// ---------------------------------------------------------------------------
// K2b: rearrange tconv_w (O,I,9,1) -> GEMM A-matrix Wt[c][dt*64+ci]
// ---------------------------------------------------------------------------
__global__ void k2_wt(const float* __restrict__ tw, float* __restrict__ wt_ws) {
  const int i = blockIdx.x * 256 + threadIdx.x;
  if (i >= 64 * 576) return;
  const int c = i / 576, r = i - c * 576;
  const int dt = r >> 6, ci = r & 63;
  wt_ws[i] = tw[(c * 64 + ci) * 9 + dt];
}

// ---------------------------------------------------------------------------
// K3: BN1+ReLU applied on load, 9-tap temporal conv as GEMM (M=64,N=32,K=576).
// Writes h (+tconv bias) straight into d_out and BN2 partials to pbuf2.
// ---------------------------------------------------------------------------
__global__ __launch_bounds__(256)
void k3_tconv(const float* __restrict__ g_ws, const float* __restrict__ wt_ws,
              const float* __restrict__ stats1, const float* __restrict__ tb,
              float* __restrict__ hout, float* __restrict__ pbuf2) {
  extern __shared__ float smem[];
  float* Wt  = smem;                 // [64][578]
  float* R   = Wt + 64 * 578;        // [64][18][36]  (t halo, BN1+ReLU'd)
  float* s1  = R + 64 * 18 * 36;     // 64
  float* sh1 = s1 + 64;              // 64
  float* tbl = sh1 + 64;             // 64

  const int n    = blockIdx.x;
  const int t0   = blockIdx.y * CT1;
  const int tid  = threadIdx.x;
  const int wave = tid >> 5;
  const int lane = tid & 31;

  if (tid < 64) {
    s1[tid]  = stats1[tid];
    sh1[tid] = stats1[COUT + tid];
    tbl[tid] = tb[tid];
  }
  for (int i = tid; i < 64 * 576; i += 256)
    Wt[(i / 576) * 578 + (i % 576)] = wt_ws[i];
  __syncthreads();  // s1/sh1 ready

  // Stage BN1(g)+ReLU halo: tt = 0..17 covers t0-4 .. t0+CT1+3 (zero padded).
  for (int i = tid; i < 64 * 18 * 36; i += 256) {
    const int ci = i / (18 * 36);
    const int r  = i - ci * (18 * 36);
    const int tt = r / 36, v = r - tt * 36;
    float val = 0.f;
    const int tg = t0 + tt - 4;
    if (v < VV && tg >= 0 && tg < TT) {
      const float raw = g_ws[((n * COUT + ci) * TT + tg) * VV + v];
      val = fmaxf(0.f, s1[ci] * raw + sh1[ci]);
    }
    R[i] = val;
  }
  __syncthreads();

  const int ct = wave >> 1;
  const int wt = wave & 1;
  float bsum[8], bsq[8];
#pragma unroll
  for (int j = 0; j < 8; ++j) { bsum[j] = 0.f; bsq[j] = 0.f; }

  for (int t = 0; t < CT1; ++t) {
    v8f acc = {};
#pragma unroll
    for (int dt = 0; dt < 9; ++dt) {
      const float* Rbase = R + (t + dt) * 36;  // ci-row stride = 18*36
#pragma unroll
      for (int s = 0; s < 16; ++s) {
        v2f a = load_A(Wt, 578, ct * 16, dt * 64 + s * 4);
        v2f b = load_B(Rbase, 18 * 36, s * 4, wt * 16);
        acc = wmma4(a, b, acc);
      }
    }
    const int v  = wt * 16 + (lane & 15);
    const int mh = ct * 16 + ((lane >> 4) << 3);
    const int tg = t0 + t;
    if (v < VV) {
#pragma unroll
      for (int j = 0; j < 8; ++j) {
        const float val = acc[j] + tbl[mh + j];
        hout[((n * COUT + (mh + j)) * TT + tg) * VV + v] = val;
        bsum[j] += val;
        bsq[j]  += val * val;
      }
    }
  }

#pragma unroll
  for (int j = 0; j < 8; ++j) {
#pragma unroll
    for (int m = 1; m < 16; m <<= 1) {
      bsum[j] += __shfl_xor(bsum[j], m, 32);
      bsq[j]  += __shfl_xor(bsq[j], m, 32);
    }
  }
  if ((lane & 15) == 0) {
    const int wg = blockIdx.y * gridDim.x + blockIdx.x;
    const int hi = lane >> 4;
#pragma unroll
    for (int j = 0; j < 8; ++j) {
      const int cl = hi * 8 + j;
      const int base = ((wg * 8 + wave) * 16 + cl) * 2;
      pbuf2[base + 0] = bsum[j];
      pbuf2[base + 1] = bsq[j];
    }
  }
}

// ---------------------------------------------------------------------------
// K5: out = relu(bn2(h) + x) in place over d_out, plus copy A to the tail.
// (h lives in d_out; per-thread read-then-write same index -> no restrict)
// ---------------------------------------------------------------------------
__global__ void k5_final(const float* __restrict__ x, const float* h,
                         const float* __restrict__ stats2,
                         const float* __restrict__ Ag, float* out) {
  const int i = blockIdx.x * 256 + threadIdx.x;
  const int TOT = NB * COUT * TT * VV;  // 15,360,000
  if (i < TOT) {
    const int c = (i / (TT * VV)) % COUT;
    const float val = stats2[c] * h[i] + stats2[COUT + c] + x[i];
    out[i] = fmaxf(val, 0.f);
  }
  if (i < EDGE * VV * VV) out[TOT + i] = Ag[i];  // second tuple element: A
}

// ---------------------------------------------------------------------------
extern "C" void kernel_launch(void* const* d_in, const int* in_sizes, int n_in,
                              void* d_out, int out_size, void* d_ws,
                              size_t ws_size, hipStream_t stream) {
  const float* x  = (const float*)d_in[0];
  const float* Ag = (const float*)d_in[1];
  const float* Bg = (const float*)d_in[2];
  const float* cw = (const float*)d_in[3];
  const float* cb = (const float*)d_in[4];
  const float* tw = (const float*)d_in[5];
  const float* tb = (const float*)d_in[6];
  const float* g1 = (const float*)d_in[7];
  const float* b1 = (const float*)d_in[8];
  const float* g2 = (const float*)d_in[9];
  const float* b2 = (const float*)d_in[10];
  float* out = (float*)d_out;

  // Workspace carve (floats): g (61.4MB) + partials + stats + Wt  (~64MB)
  float* ws     = (float*)d_ws;
  float* g_ws   = ws;                      // 15,360,000
  float* pbuf1  = g_ws + 15360000;         // 245,760
  float* pbuf2  = pbuf1 + 245760;          // 245,760
  float* stats1 = pbuf2 + 245760;          // 128
  float* stats2 = stats1 + 128;            // 128
  float* wt_ws  = stats2 + 128;            // 36,864

  const size_t lds1 =
      (size_t)(384 * 66 + 384 * 36 + 6 * 28 * 40 + 64 * 40 + 384) * 4;  // ~195KB
  const size_t lds3 = (size_t)(64 * 578 + 64 * 18 * 36 + 192) * 4;      // ~315KB
  (void)hipFuncSetAttribute((const void*)k1_gcn,
                            hipFuncAttributeMaxDynamicSharedMemorySize,
                            (int)lds1);
  (void)hipFuncSetAttribute((const void*)k3_tconv,
                            hipFuncAttributeMaxDynamicSharedMemorySize,
                            (int)lds3);

  dim3 grid(NB, NCHUNK);  // 32 x 30 = 960 workgroups of 8 waves
  k1_gcn<<<grid, 256, lds1, stream>>>(x, Ag, Bg, cw, cb, g_ws, pbuf1);
  k2_stats<<<1, 64, 0, stream>>>(pbuf1, g1, b1, stats1);
  k2_wt<<<(64 * 576 + 255) / 256, 256, 0, stream>>>(tw, wt_ws);
  k3_tconv<<<grid, 256, lds3, stream>>>(g_ws, wt_ws, stats1, tb, out, pbuf2);
  k2_stats<<<1, 64, 0, stream>>>(pbuf2, g2, b2, stats2);
  const int TOT = NB * COUT * TT * VV;
  k5_final<<<(TOT + 255) / 256, 256, 0, stream>>>(x, out, stats2, Ag, out);
}